// LinearAttention_17377437679936
// MI455X (gfx1250) — compile-verified
//
#include <hip/hip_runtime.h>

// Linear attention (ELU+1 feature map), fp32, N=8 L=S=8192 H=8 D=32.
// Bandwidth-bound (~268MB @ 23.3TB/s); fp32 WMMA 16x16x4 keeps exact precision
// while putting all GEMM work on the matrix pipe.

#define N_ 8
#define L_ 8192
#define S_ 8192
#define H_ 8
#define D_ 32
#define RS (H_ * D_)            // 256 floats between consecutive s/l rows
#define NH (N_ * H_)            // 64
#define CHUNKS 64
#define CHUNK_ROWS (S_ / CHUNKS) // 128
#define SLOT 1056               // 32*32 KV + 32 Ksum floats per slot

typedef __attribute__((ext_vector_type(2))) float v2f;
typedef __attribute__((ext_vector_type(8))) float v8f;

__device__ __forceinline__ float elu1(float x) {
  // elu(x)+1 : x>0 ? x+1 : exp(x)
  return x > 0.f ? x + 1.f : __expf(x);
}

__device__ __forceinline__ v8f wmma_f32(v2f a, v2f b, v8f c) {
  // (neg_a, A, neg_b, B, c_mod, C, reuse_a, reuse_b)
  return __builtin_amdgcn_wmma_f32_16x16x4_f32(false, a, false, b, (short)0, c,
                                               false, false);
}

// ---------------- Pass 1: partial KV (32x32) + partial Ksum per chunk -------
__global__ void __launch_bounds__(256)
kv_partial_kernel(const float* __restrict__ Kp, const float* __restrict__ Vp,
                  float* __restrict__ part) {
  const int wave = threadIdx.x >> 5;
  const int lane = threadIdx.x & 31;
  const int g    = blockIdx.x * 8 + wave;   // 0 .. NH*CHUNKS-1
  const int nh   = g / CHUNKS;
  const int ck   = g % CHUNKS;
  const int n    = nh / H_;
  const int h    = nh % H_;
  const int col  = lane & 15;
  const int half = lane >> 4;               // 0: k%4 in {0,1}; 1: {2,3}
  const float invS = 1.0f / (float)S_;

  const size_t base = ((size_t)n * S_) * RS + (size_t)h * D_;
  const int s0 = ck * CHUNK_ROWS;

  v8f c00 = {}, c01 = {}, c10 = {}, c11 = {};
  float ks0 = 0.f, ks1 = 0.f;

  for (int s = s0; s < s0 + CHUNK_ROWS; s += 4) {
    const int rA = s + 2 * half;
    const float* k0 = Kp + base + (size_t)rA * RS;
    const float* k1 = k0 + RS;
    const float* v0 = Vp + base + (size_t)rA * RS;
    const float* v1 = v0 + RS;
    v2f a0, a1, b0, b1;
    float t;
    // A tiles: A[m=d][k] = fm(K[s+k][d]); lane holds m=col, k=2*half+{0,1}
    t = elu1(k0[col]);      a0.x = t; ks0 += t;
    t = elu1(k1[col]);      a0.y = t; ks0 += t;
    t = elu1(k0[col + 16]); a1.x = t; ks1 += t;
    t = elu1(k1[col + 16]); a1.y = t; ks1 += t;
    // B tiles: B[k][e] = V[s+k][e]/S; lane holds n=col, k=2*half+{0,1}
    b0.x = v0[col]      * invS; b0.y = v1[col]      * invS;
    b1.x = v0[col + 16] * invS; b1.y = v1[col + 16] * invS;
    c00 = wmma_f32(a0, b0, c00);
    c01 = wmma_f32(a0, b1, c01);
    c10 = wmma_f32(a1, b0, c10);
    c11 = wmma_f32(a1, b1, c11);
  }
  // lanes l and l+16 each covered half of the rows for columns col / col+16
  ks0 += __shfl_xor(ks0, 16, 32);
  ks1 += __shfl_xor(ks1, 16, 32);

  float* outp = part + (size_t)g * SLOT;
#pragma unroll
  for (int r = 0; r < 8; ++r) {
    const int m = r + 8 * half;            // C/D layout: vgpr=m&7, lane half
    outp[m * 32 + col]              = c00[r];
    outp[m * 32 + col + 16]         = c01[r];
    outp[(16 + m) * 32 + col]       = c10[r];
    outp[(16 + m) * 32 + col + 16]  = c11[r];
  }
  if (lane < 16) {
    outp[1024 + col]      = ks0;
    outp[1024 + col + 16] = ks1;
  }
}

// ---------------- Pass 2: deterministic reduction over chunks ---------------
__global__ void __launch_bounds__(256)
kv_reduce_kernel(const float* __restrict__ part, float* __restrict__ kv) {
  const int nh = blockIdx.x;
  for (int e = threadIdx.x; e < SLOT; e += 256) {
    float acc = 0.f;
    const float* p = part + (size_t)nh * CHUNKS * SLOT + e;
#pragma unroll 4
    for (int c = 0; c < CHUNKS; ++c) acc += p[(size_t)c * SLOT];
    kv[(size_t)nh * SLOT + e] = acc;
  }
}

// ---------------- Pass 3: out = fm(Q) @ KV, scaled by Z*S -------------------
__global__ void __launch_bounds__(256)
attn_out_kernel(const float* __restrict__ Qp, const float* __restrict__ kvbuf,
                float* __restrict__ outp) {
  __shared__ float sKV[1024];
  __shared__ float sKs[32];
  const int tilesPerNH = L_ / 128;          // 64 blocks of 128 rows per (n,h)
  const int nh = blockIdx.x / tilesPerNH;
  const int tb = blockIdx.x % tilesPerNH;
  const int n  = nh / H_;
  const int h  = nh % H_;
  for (int e = threadIdx.x; e < SLOT; e += 256) {
    float v = kvbuf[(size_t)nh * SLOT + e];
    if (e < 1024) sKV[e] = v; else sKs[e - 1024] = v;
  }
  __syncthreads();

  const int wave = threadIdx.x >> 5;
  const int lane = threadIdx.x & 31;
  const int col  = lane & 15;
  const int half = lane >> 4;
  const int row0 = tb * 128 + wave * 16;
  const size_t qbase = ((size_t)n * L_ + row0) * RS + (size_t)h * D_;

  v8f c0 = {}, c1 = {};
  float dot = 0.f;
#pragma unroll
  for (int kb = 0; kb < 32; kb += 4) {
    const int kk = kb + 2 * half;
    // A: lane holds row m=col, k = kk,kk+1 (contiguous in the Q row -> v2f)
    v2f q = *(const v2f*)(Qp + qbase + (size_t)col * RS + kk);
    v2f a;
    float t0 = elu1(q.x), t1 = elu1(q.y);
    a.x = t0; a.y = t1;
    dot += t0 * sKs[kk] + t1 * sKs[kk + 1];
    v2f b0, b1;
    b0.x = sKV[kk * 32 + col];            b0.y = sKV[(kk + 1) * 32 + col];
    b1.x = sKV[kk * 32 + col + 16];       b1.y = sKV[(kk + 1) * 32 + col + 16];
    c0 = wmma_f32(a, b0, c0);
    c1 = wmma_f32(a, b1, c1);
  }
  // lanes l / l+16 covered complementary k's for row col
  dot += __shfl_xor(dot, 16, 32);
  const float z = (float)S_ / (dot + 1e-6f);   // Z * v_length
#pragma unroll
  for (int r = 0; r < 8; ++r) {
    const int m = r + 8 * half;
    const float zr = __shfl(z, m, 32);         // z lives in lane == row index
    float* orow = outp + ((size_t)n * L_ + row0 + m) * RS + (size_t)h * D_;
    orow[col]      = c0[r] * zr;
    orow[col + 16] = c1[r] * zr;
  }
}

extern "C" void kernel_launch(void* const* d_in, const int* in_sizes, int n_in,
                              void* d_out, int out_size, void* d_ws,
                              size_t ws_size, hipStream_t stream) {
  const float* Q = (const float*)d_in[0];
  const float* K = (const float*)d_in[1];
  const float* V = (const float*)d_in[2];
  float* out = (float*)d_out;

  float* part    = (float*)d_ws;                         // NH*CHUNKS*SLOT
  float* kvfinal = part + (size_t)NH * CHUNKS * SLOT;    // NH*SLOT

  kv_partial_kernel<<<NH * CHUNKS / 8, 256, 0, stream>>>(K, V, part);
  kv_reduce_kernel<<<NH, 256, 0, stream>>>(part, kvfinal);
  attn_out_kernel<<<NH * (L_ / 128), 256, 0, stream>>>(Q, kvfinal, out);
}